// GCN_66649302499531
// MI455X (gfx1250) — compile-verified
//
#include <hip/hip_runtime.h>

// Problem constants (match reference)
#define NN   20000
#define GG   400
#define MAXM 256      // max members per group (Binomial(20000,1/400): mean 50, ~29 sigma headroom)

typedef __attribute__((ext_vector_type(2))) float v2f;
typedef __attribute__((ext_vector_type(8))) float v8f;

// ---------------------------------------------------------------------------
// Stage 1: build ordered per-group member lists.
// One block per group; block-wide inclusive scan appends matching node ids in
// increasing order, so position p == rank == (#earlier same-group nodes).
// ---------------------------------------------------------------------------
__global__ __launch_bounds__(256)
void build_groups_kernel(const int* __restrict__ idx,
                         int* __restrict__ members,
                         int* __restrict__ counts, int n)
{
    const int g   = blockIdx.x;
    const int tid = threadIdx.x;
    __shared__ int flags[256];
    __shared__ int base_s;
    if (tid == 0) base_s = 0;
    __syncthreads();

    for (int start = 0; start < n; start += 256) {
        const int i = start + tid;
        const int f = (i < n && idx[i] == g) ? 1 : 0;
        flags[tid] = f;
        __syncthreads();
        for (int off = 1; off < 256; off <<= 1) {   // inclusive Hillis-Steele scan
            const int v   = flags[tid];
            const int add = (tid >= off) ? flags[tid - off] : 0;
            __syncthreads();
            flags[tid] = v + add;
            __syncthreads();
        }
        const int incl = flags[tid];
        const int tot  = flags[255];
        const int pos  = base_s + (incl - f);
        if (f && pos < MAXM) members[g * MAXM + pos] = i;
        __syncthreads();
        if (tid == 0) base_s += tot;
        __syncthreads();
    }
    if (tid == 0) counts[g] = (base_s < MAXM) ? base_s : MAXM;
}

// ---------------------------------------------------------------------------
// Stage 2: fp32 WMMA GEMM  out[N,F] = A[N,K] @ W[K,F]
// Block = 32x8 (8 waves). Each wave computes a 16x64 output strip using FOUR
// v8f accumulators, so one A fragment (global_load_b64) feeds 4 WMMAs and
// per-row A re-reads drop to F/64. B is staged through LDS in K-chunks of 128
// (32 KB), packed [k/2][64 cols][2] so each lane's K-pair is an aligned
// ds_load_b64 (pairable into ds_load_2addr_b64), bank-conflict free.
//
// A 16x4 layout : lane l -> m = l&15, kb = 2*(l>>4); v2f = {A[m][k+kb], A[m][k+kb+1]}
// B 4x16 layout : lane l -> n = l&15, kb = 2*(l>>4); v2f = {B[k+kb][n], B[k+kb+1][n]}
// C/D 16x16     : lane l -> n = l&15; VGPR j -> m = j + 8*(l>>4)
//
// No early return: out-of-range waves clamp their A row to 0 and skip the
// store, so barriers stay uniform and EXEC is all-1s for every WMMA.
// ---------------------------------------------------------------------------
template<int K, int F>
__global__ __launch_bounds__(256)
void gemm_wmma_kernel(const float* __restrict__ A,
                      const float* __restrict__ W,
                      float* __restrict__ out, int n)
{
    constexpr int KC = 128;                    // K staging chunk
    __shared__ float Bs[(KC / 2) * 128];       // 32 KB: [KC/2][64][2] packed pairs

    const int tid  = threadIdx.y * 32 + threadIdx.x;
    const int col0 = blockIdx.y * 64;

    const int  m_tile = blockIdx.x * 8 + threadIdx.y;
    const bool valid  = (m_tile * 16 < n);

    const int lane = threadIdx.x;
    const int half = lane >> 4;                // 0 or 1
    const int l16  = lane & 15;
    const int kb   = half * 2;

    const int    mrow = valid ? (m_tile * 16 + l16) : l16;
    const float* arow = A + (size_t)mrow * K;
    __builtin_prefetch(arow, 0, 0);

    v8f c0 = {0.f,0.f,0.f,0.f,0.f,0.f,0.f,0.f};
    v8f c1 = c0, c2 = c0, c3 = c0;

    for (int kc = 0; kc < K; kc += KC) {
        // stage B chunk: W[kc+k][col0+c], k in [0,KC), c in [0,64)
        for (int i = tid; i < KC * 64; i += 256) {
            const int k = i >> 6;
            const int cc = i & 63;
            Bs[(k >> 1) * 128 + cc * 2 + (k & 1)] = W[(size_t)(kc + k) * F + col0 + cc];
        }
        __syncthreads();

#pragma unroll 4
        for (int k = 0; k < KC; k += 4) {
            const v2f a = *(const v2f*)(arow + kc + k + kb);            // global_load_b64
            const float* bbase = &Bs[((k + kb) >> 1) * 128 + l16 * 2];
            const v2f b0 = *(const v2f*)(bbase);                         // ds_load_b64 x4
            const v2f b1 = *(const v2f*)(bbase + 32);
            const v2f b2 = *(const v2f*)(bbase + 64);
            const v2f b3 = *(const v2f*)(bbase + 96);
            c0 = __builtin_amdgcn_wmma_f32_16x16x4_f32(false, a, false, b0, (short)0, c0, false, false);
            c1 = __builtin_amdgcn_wmma_f32_16x16x4_f32(false, a, false, b1, (short)0, c1, false, false);
            c2 = __builtin_amdgcn_wmma_f32_16x16x4_f32(false, a, false, b2, (short)0, c2, false, false);
            c3 = __builtin_amdgcn_wmma_f32_16x16x4_f32(false, a, false, b3, (short)0, c3, false, false);
        }
        __syncthreads();
    }

    if (valid) {
        float* obase = out + (size_t)(m_tile * 16 + half * 8) * F + col0 + l16;
#pragma unroll
        for (int j = 0; j < 8; ++j) {
            obase[(size_t)j * F +  0] = c0[j];
            obase[(size_t)j * F + 16] = c1[j];
            obase[(size_t)j * F + 32] = c2[j];
            obase[(size_t)j * F + 48] = c3[j];
        }
    }
}

// ---------------------------------------------------------------------------
// Stage 3: per-group scaled prefix sum (the whole GCN message passing),
// + bias + ReLU; optionally fused LayerNorm (last layer) writing d_out.
// One block per group, F threads (one per feature).
//   out[c] = relu( dis[c] * cumsum_{p<=rank(c)} dis[p]*g[p] + bias )
//   dis at rank p = rsqrt(p+1)   (degree = rank + 1, incl. self-loop)
// ---------------------------------------------------------------------------
template<int F, bool DO_LN>
__global__ void agg_kernel(const float* __restrict__ gin,
                           const float* __restrict__ bias,
                           float* __restrict__ hout,
                           const int* __restrict__ members,
                           const int* __restrict__ counts,
                           const float* __restrict__ gamma,
                           const float* __restrict__ beta)
{
    const int g   = blockIdx.x;
    const int tid = threadIdx.x;
    __shared__ float red[F];

    const int   cnt = counts[g];
    const float b   = bias[tid];
    const float gm  = DO_LN ? gamma[tid] : 0.f;
    const float bt  = DO_LN ? beta[tid]  : 0.f;

    float acc = 0.f;
    for (int p = 0; p < cnt; ++p) {
        const int   node = members[g * MAXM + p];
        const float d    = rsqrtf((float)(p + 1));
        const float v    = gin[(size_t)node * F + tid];
        acc = fmaf(d, v, acc);
        float o = fmaxf(fmaf(d, acc, b), 0.f);

        if (DO_LN) {
            red[tid] = o;
            __syncthreads();
            for (int s = F / 2; s > 0; s >>= 1) {
                if (tid < s) red[tid] += red[tid + s];
                __syncthreads();
            }
            const float mu = red[0] * (1.0f / F);
            __syncthreads();
            const float dv = o - mu;
            red[tid] = dv * dv;
            __syncthreads();
            for (int s = F / 2; s > 0; s >>= 1) {
                if (tid < s) red[tid] += red[tid + s];
                __syncthreads();
            }
            const float var = red[0] * (1.0f / F);
            __syncthreads();
            hout[(size_t)node * F + tid] = dv * rsqrtf(var + 1e-5f) * gm + bt;
        } else {
            hout[(size_t)node * F + tid] = o;
        }
    }
}

// ---------------------------------------------------------------------------
// Launch
// ---------------------------------------------------------------------------
extern "C" void kernel_launch(void* const* d_in, const int* in_sizes, int n_in,
                              void* d_out, int out_size, void* d_ws, size_t ws_size,
                              hipStream_t stream)
{
    const float* x     = (const float*)d_in[0];
    const int*   idx   = (const int*)  d_in[1];   // int32 on device (JAX default x32)
    const float* W1    = (const float*)d_in[2];
    const float* b1    = (const float*)d_in[3];
    const float* W2    = (const float*)d_in[4];
    const float* b2    = (const float*)d_in[5];
    const float* W3    = (const float*)d_in[6];
    const float* b3    = (const float*)d_in[7];
    const float* gamma = (const float*)d_in[8];
    const float* beta  = (const float*)d_in[9];
    float* out = (float*)d_out;

    char* ws = (char*)d_ws;
    int*   counts  = (int*)ws;                        //  1,600 B
    int*   members = (int*)(ws + 4096);               //  409,600 B
    float* h0      = (float*)(ws + 524288);           //  N*256 f32 = 20.48 MB
    float* h1      = h0 + (size_t)NN * 256;           //  N*256 f32 = 20.48 MB

    build_groups_kernel<<<GG, 256, 0, stream>>>(idx, members, counts, NN);

    const dim3 blk(32, 8);
    const int  mblocks = (NN / 16 + 7) / 8;           // 157

    // layer 1: h0 = x @ W1 ; h1 = relu(agg(h0) + b1)
    gemm_wmma_kernel<128, 256><<<dim3(mblocks, 256 / 64), blk, 0, stream>>>(x, W1, h0, NN);
    agg_kernel<256, false><<<GG, 256, 0, stream>>>(h0, b1, h1, members, counts, nullptr, nullptr);

    // layer 2
    gemm_wmma_kernel<256, 256><<<dim3(mblocks, 256 / 64), blk, 0, stream>>>(h1, W2, h0, NN);
    agg_kernel<256, false><<<GG, 256, 0, stream>>>(h0, b2, h1, members, counts, nullptr, nullptr);

    // layer 3 + fused LayerNorm -> d_out
    gemm_wmma_kernel<256, 128><<<dim3(mblocks, 128 / 64), blk, 0, stream>>>(h1, W3, h0, NN);
    agg_kernel<128, true><<<GG, 128, 0, stream>>>(h0, b3, out, members, counts, gamma, beta);
}